// GVMessage_88545045775086
// MI455X (gfx1250) — compile-verified
//
#include <hip/hip_runtime.h>
#include <math.h>

#define BN 2
#define NN 384
#define NTOK (BN*NN)          // 768
#define NTILE (NN/16)         // 24 j-tiles per row

typedef float v2f __attribute__((ext_vector_type(2)));
typedef float v8f __attribute__((ext_vector_type(8)));

static __device__ __forceinline__ v8f wmma_f32(v2f a, v2f b, v8f c) {
  // D = A(16x4) * B(4x16) + C(16x16), all f32, wave32
  return __builtin_amdgcn_wmma_f32_16x16x4_f32(false, a, false, b, (short)0, c, false, false);
}

static __device__ __forceinline__ float sigmoidf(float x) {
  return 1.0f / (1.0f + expf(-x));
}

// ---------------------------------------------------------------------------
// Kernel A: node QKV GVL. One block (128 thr) per token.
// ---------------------------------------------------------------------------
__global__ void node_qkv_kernel(const float* __restrict__ x_sca, const float* __restrict__ x_vec,
    const float* __restrict__ W1, const float* __restrict__ W2, const float* __restrict__ Wg,
    const float* __restrict__ bg, const float* __restrict__ Ws, const float* __restrict__ bs,
    float* __restrict__ sca_qkv, float* __restrict__ vec_qkv)
{
  int tok = blockIdx.x;
  int tid = threadIdx.x;
  __shared__ float sx[192];    // x_vec [d*3+c]
  __shared__ float sV1[576];   // v1 [o*3+c]
  __shared__ float cat[320];   // [v1n(192), sca(128)]
  __shared__ float sSO[384];
  __shared__ float sG[192];
  const float* xv = x_vec + (long)tok * 192;
  const float* xs = x_sca + (long)tok * 128;
  for (int p = tid; p < 192; p += 128) sx[p] = xv[p];
  if (tid < 128) cat[192 + tid] = xs[tid];
  __syncthreads();
  for (int p = tid; p < 576; p += 128) {
    int o = p / 3, c = p - o * 3;
    float acc = 0.f;
    for (int d = 0; d < 64; ++d) acc += W1[o * 64 + d] * sx[d * 3 + c];
    sV1[p] = acc;
  }
  __syncthreads();
  for (int o = tid; o < 192; o += 128) {
    float a0 = sV1[o*3], a1 = sV1[o*3+1], a2 = sV1[o*3+2];
    cat[o] = sqrtf(a0*a0 + a1*a1 + a2*a2);
  }
  __syncthreads();
  for (int o = tid; o < 384; o += 128) {
    float acc = bs[o];
    for (int k = 0; k < 320; ++k) acc += cat[k] * Ws[o * 320 + k];
    sSO[o] = acc;
    sca_qkv[(long)tok * 384 + o] = acc;
  }
  __syncthreads();
  for (int o = tid; o < 192; o += 128) {
    float acc = bg[o];
    for (int k = 0; k < 384; ++k) acc += sSO[k] * Wg[o * 384 + k];
    sG[o] = sigmoidf(acc);
  }
  __syncthreads();
  for (int p = tid; p < 576; p += 128) {
    int o = p / 3, c = p - o * 3;
    float acc = 0.f;
    for (int d = 0; d < 192; ++d) acc += W2[o * 192 + d] * sV1[d * 3 + c];
    vec_qkv[(long)tok * 576 + p] = sG[o] * acc;
  }
}

// ---------------------------------------------------------------------------
// Kernel B: fused edge GVL + attention logits. One wave per 16-edge tile.
// All four edge GEMMs run on V_WMMA_F32_16X16X4_F32; es/ev never hit HBM.
// ---------------------------------------------------------------------------
__global__ void edge_att_kernel(const float* __restrict__ edge_sca, const float* __restrict__ edge_vec,
    const float* __restrict__ sca_qkv, const float* __restrict__ vec_qkv,
    const float* __restrict__ W1, const float* __restrict__ W2, const float* __restrict__ Wg,
    const float* __restrict__ bg, const float* __restrict__ Ws, const float* __restrict__ bs,
    float* __restrict__ att_s, float* __restrict__ att_v)
{
  __shared__ float sEV[16 * 97];   // edge_vec tile  [t][d*3+c], padded
  __shared__ float sCat[16 * 129]; // [v1n(64), edge_sca(64)] per token, padded
  __shared__ float sV1[16 * 195];  // v1 [t][c*65 + o], padded
  __shared__ float sES[16 * 129];  // es  [t][f], padded

  int lane = threadIdx.x;          // wave32
  int lr = lane & 15, lh = lane >> 4;
  int tile = blockIdx.x;           // 0 .. BN*NN*NTILE-1
  int jt = tile % NTILE;
  int bi = tile / NTILE;           // b*NN + i
  int b  = bi / NN;
  int j0 = jt * 16;
  long edge0 = (long)bi * NN + j0; // flat (b,i,j0) edge index

  const float* evp = edge_vec + edge0 * 96;
  const float* esp = edge_sca + edge0 * 64;

  // ---- stage edge tile into LDS (coalesced float4) ----
  for (int k = lane; k < 384; k += 32) {             // 16*96 floats
    float4 v = ((const float4*)evp)[k];
    int idx = k * 4; int t = idx / 96; int r = idx - t * 96;
    float* dst = &sEV[t * 97 + r];
    dst[0] = v.x; dst[1] = v.y; dst[2] = v.z; dst[3] = v.w;
  }
  for (int k = lane; k < 256; k += 32) {             // 16*64 floats
    float4 v = ((const float4*)esp)[k];
    int idx = k * 4; int t = idx >> 6; int r = idx & 63;
    float* dst = &sCat[t * 129 + 64 + r];
    dst[0] = v.x; dst[1] = v.y; dst[2] = v.z; dst[3] = v.w;
  }
  __syncthreads();

  // ---- Stage 1: v1 = W1(64x32) @ ev, per coord; also v1 norms into cat ----
  for (int nt = 0; nt < 4; ++nt) {
    int n0 = nt * 16;
    v8f acc[3];
    #pragma unroll
    for (int c = 0; c < 3; ++c) {
      v8f a = {0,0,0,0,0,0,0,0};
      #pragma unroll
      for (int k0 = 0; k0 < 32; k0 += 4) {
        int ka = k0 + lh * 2;
        v2f A; A.x = sEV[lr * 97 + ka * 3 + c];      A.y = sEV[lr * 97 + (ka + 1) * 3 + c];
        v2f Bf; Bf.x = W1[(n0 + lr) * 32 + ka];      Bf.y = W1[(n0 + lr) * 32 + ka + 1];
        a = wmma_f32(A, Bf, a);
      }
      acc[c] = a;
    }
    #pragma unroll
    for (int v = 0; v < 8; ++v) {
      int t = v + lh * 8;
      float vx = acc[0][v], vy = acc[1][v], vz = acc[2][v];
      sV1[t * 195 + 0 * 65 + n0 + lr] = vx;
      sV1[t * 195 + 1 * 65 + n0 + lr] = vy;
      sV1[t * 195 + 2 * 65 + n0 + lr] = vz;
      sCat[t * 129 + n0 + lr] = sqrtf(vx * vx + vy * vy + vz * vz);
    }
  }
  __syncthreads();

  // ---- Stage 2: es = [v1n, sca] @ Ws(128x128)^T + bs ----
  for (int nt = 0; nt < 8; ++nt) {
    int n0 = nt * 16;
    v8f a = {0,0,0,0,0,0,0,0};
    for (int k0 = 0; k0 < 128; k0 += 4) {
      int ka = k0 + lh * 2;
      v2f A; A.x = sCat[lr * 129 + ka];              A.y = sCat[lr * 129 + ka + 1];
      v2f Bf; Bf.x = Ws[(n0 + lr) * 128 + ka];       Bf.y = Ws[(n0 + lr) * 128 + ka + 1];
      a = wmma_f32(A, Bf, a);
    }
    float bias = bs[n0 + lr];
    #pragma unroll
    for (int v = 0; v < 8; ++v) sES[(v + lh * 8) * 129 + n0 + lr] = a[v] + bias;
  }
  __syncthreads();

  // ---- scalar logits: sum_d sq*sk*es / sqrt(16) ----
  const float* sqp_base = sca_qkv + (long)bi * 384;
  for (int p = lane; p < 128; p += 32) {
    int t = p >> 3, h = p & 7;
    const float* sqp = sqp_base + h * 16;
    const float* skp = sca_qkv + ((long)b * NN + j0 + t) * 384 + 128 + h * 16;
    float s = 0.f;
    #pragma unroll
    for (int d = 0; d < 16; ++d) s += sES[t * 129 + h * 16 + d] * sqp[d] * skp[d];
    att_s[(edge0 + t) * 8 + h] = s * 0.25f;
  }

  // ---- Stage 3+4: g = sigmoid(es@Wg^T+bg); v2 = W2@v1; ev=g*v2 -> v-logits ----
  const float* vqb = vec_qkv + (long)bi * 576;
  for (int nt = 0; nt < 4; ++nt) {
    int n0 = nt * 16;
    v8f ag = {0,0,0,0,0,0,0,0};
    for (int k0 = 0; k0 < 128; k0 += 4) {
      int ka = k0 + lh * 2;
      v2f A; A.x = sES[lr * 129 + ka];               A.y = sES[lr * 129 + ka + 1];
      v2f Bf; Bf.x = Wg[(n0 + lr) * 128 + ka];       Bf.y = Wg[(n0 + lr) * 128 + ka + 1];
      ag = wmma_f32(A, Bf, ag);
    }
    float bgv = bg[n0 + lr];
    v8f av[3];
    #pragma unroll
    for (int c = 0; c < 3; ++c) {
      v8f a = {0,0,0,0,0,0,0,0};
      #pragma unroll
      for (int k0 = 0; k0 < 64; k0 += 4) {
        int ka = k0 + lh * 2;
        v2f A; A.x = sV1[lr * 195 + c * 65 + ka];    A.y = sV1[lr * 195 + c * 65 + ka + 1];
        v2f Bf; Bf.x = W2[(n0 + lr) * 64 + ka];      Bf.y = W2[(n0 + lr) * 64 + ka + 1];
        a = wmma_f32(A, Bf, a);
      }
      av[c] = a;
    }
    int oo = n0 + lr;                 // vector channel 0..63
    int h = oo >> 3, dd = oo & 7;     // head, d-within-head
    float vq0 = vqb[oo * 3 + 0], vq1 = vqb[oo * 3 + 1], vq2 = vqb[oo * 3 + 2];
    #pragma unroll
    for (int v = 0; v < 8; ++v) {
      int t = v + lh * 8;
      float g = sigmoidf(ag[v] + bgv);
      const float* vkp = vec_qkv + ((long)b * NN + j0 + t) * 576 + (64 + oo) * 3;
      float part = g * (av[0][v] * vq0 * vkp[0] + av[1][v] * vq1 * vkp[1] + av[2][v] * vq2 * vkp[2]);
      part += __shfl_xor(part, 1, 32);
      part += __shfl_xor(part, 2, 32);
      part += __shfl_xor(part, 4, 32);
      if (dd == 0)
        att_v[(edge0 + t) * 8 + h] = part * 0.20412414523193154f; // 1/sqrt(24)
    }
  }
}

// ---------------------------------------------------------------------------
// Kernel C: masked softmax over j (in place), both logit tensors.
// ---------------------------------------------------------------------------
__global__ void softmax_kernel(float* __restrict__ att_s, float* __restrict__ att_v,
                               const int* __restrict__ mask)
{
  int bi = blockIdx.x;
  int b = bi / NN;
  int tid = threadIdx.x;              // 128
  int h = tid & 7, jj = tid >> 3;     // jj 0..15
  __shared__ float red[128];
  __shared__ float fmax[8];
  __shared__ float fsum[8];
  for (int a = 0; a < 2; ++a) {
    float* row = (a == 0 ? att_s : att_v) + (long)bi * NN * 8;
    float m = -3.402823466e38f;
    for (int j = jj; j < NN; j += 16)
      if (mask[b * NN + j]) m = fmaxf(m, row[j * 8 + h]);
    red[tid] = m;
    __syncthreads();
    if (tid < 8) {
      float mm = red[tid];
      for (int k = 1; k < 16; ++k) mm = fmaxf(mm, red[tid + k * 8]);
      fmax[tid] = mm;
    }
    __syncthreads();
    float mm = fmax[h];
    float s = 0.f;
    for (int j = jj; j < NN; j += 16)
      if (mask[b * NN + j]) s += expf(row[j * 8 + h] - mm);
    red[tid] = s;
    __syncthreads();
    if (tid < 8) {
      float ss = red[tid];
      for (int k = 1; k < 16; ++k) ss += red[tid + k * 8];
      fsum[tid] = ss;
    }
    __syncthreads();
    float inv = 1.f / fsum[h];
    for (int j = jj; j < NN; j += 16)
      row[j * 8 + h] = mask[b * NN + j] ? expf(row[j * 8 + h] - mm) * inv : 0.f;
    __syncthreads();
  }
}

// ---------------------------------------------------------------------------
// Kernel D: apply attention: so = att_s @ sv ; vo = att_v @ vv.
// ---------------------------------------------------------------------------
__global__ void apply_att_kernel(const float* __restrict__ att_s, const float* __restrict__ att_v,
    const float* __restrict__ sca_qkv, const float* __restrict__ vec_qkv,
    float* __restrict__ so_mid, float* __restrict__ vo_mid)
{
  int bi = blockIdx.x; int b = bi / NN;
  int tid = threadIdx.x;              // 128
  __shared__ float rs[NN * 8];
  __shared__ float rv[NN * 8];
  const float* as = att_s + (long)bi * NN * 8;
  const float* av = att_v + (long)bi * NN * 8;
  for (int p = tid; p < NN * 8; p += 128) { rs[p] = as[p]; rv[p] = av[p]; }
  __syncthreads();
  {
    int h = tid >> 4, d = tid & 15;   // 128 scalar outputs
    float acc = 0.f;
    for (int j = 0; j < NN; ++j)
      acc += rs[j * 8 + h] * sca_qkv[((long)b * NN + j) * 384 + 256 + h * 16 + d];
    so_mid[(long)bi * 128 + tid] = acc;
  }
  for (int p = tid; p < 192; p += 128) {
    int o = p / 3, c = p - o * 3;
    int h = o >> 3;
    float acc = 0.f;
    for (int j = 0; j < NN; ++j)
      acc += rv[j * 8 + h] * vec_qkv[((long)b * NN + j) * 576 + (128 + o) * 3 + c];
    vo_mid[(long)bi * 192 + p] = acc;
  }
}

// ---------------------------------------------------------------------------
// Kernel E: output GVL + vnl_leaky + leaky_relu -> d_out.
// ---------------------------------------------------------------------------
__global__ void out_kernel(const float* __restrict__ so_mid, const float* __restrict__ vo_mid,
    const float* __restrict__ W1, const float* __restrict__ W2, const float* __restrict__ Wg,
    const float* __restrict__ bg, const float* __restrict__ Ws, const float* __restrict__ bs,
    const float* __restrict__ actW, float* __restrict__ out)
{
  int tok = blockIdx.x; int tid = threadIdx.x; // 128
  __shared__ float sv0[192];  // vo_mid [o*3+c]
  __shared__ float sV1[192];
  __shared__ float cat[192];  // [v1n(64), so(128)]
  __shared__ float sSO[128];
  __shared__ float sG[64];
  __shared__ float sVo[192];
  __shared__ float sVa[192];
  for (int p = tid; p < 192; p += 128) sv0[p] = vo_mid[(long)tok * 192 + p];
  if (tid < 128) cat[64 + tid] = so_mid[(long)tok * 128 + tid];
  __syncthreads();
  for (int p = tid; p < 192; p += 128) {
    int o = p / 3, c = p - o * 3;
    float acc = 0.f;
    for (int d = 0; d < 64; ++d) acc += W1[o * 64 + d] * sv0[d * 3 + c];
    sV1[p] = acc;
  }
  __syncthreads();
  if (tid < 64) {
    float a0 = sV1[tid*3], a1 = sV1[tid*3+1], a2 = sV1[tid*3+2];
    cat[tid] = sqrtf(a0*a0 + a1*a1 + a2*a2);
  }
  __syncthreads();
  {
    float acc = bs[tid];
    for (int k = 0; k < 192; ++k) acc += cat[k] * Ws[tid * 192 + k];
    sSO[tid] = acc;
  }
  __syncthreads();
  if (tid < 64) {
    float acc = bg[tid];
    for (int k = 0; k < 128; ++k) acc += sSO[k] * Wg[tid * 128 + k];
    sG[tid] = sigmoidf(acc);
  }
  __syncthreads();
  for (int p = tid; p < 192; p += 128) {
    int o = p / 3, c = p - o * 3;
    float acc = 0.f;
    for (int d = 0; d < 64; ++d) acc += W2[o * 64 + d] * sV1[d * 3 + c];
    sVo[p] = sG[o] * acc;
  }
  __syncthreads();
  for (int p = tid; p < 192; p += 128) {
    int o = p / 3, c = p - o * 3;
    float acc = 0.f;
    for (int d = 0; d < 64; ++d) acc += actW[o * 64 + d] * sVo[d * 3 + c];
    sVa[p] = acc;
  }
  __syncthreads();
  {
    float x = sSO[tid];
    out[(long)tok * 128 + tid] = x >= 0.f ? x : 0.01f * x;   // leaky_relu
  }
  if (tid < 64) {
    int o = tid;
    float v0 = sVo[o*3], v1 = sVo[o*3+1], v2 = sVo[o*3+2];
    float a0 = sVa[o*3], a1 = sVa[o*3+1], a2 = sVa[o*3+2];
    float dot = v0*a0 + v1*a1 + v2*a2;
    float nsq = a0*a0 + a1*a1 + a2*a2;
    float r0 = v0, r1 = v1, r2 = v2;
    if (dot < 0.f) {
      float f = 0.99f * dot / (nsq + 1e-8f);
      r0 = v0 - a0 * f; r1 = v1 - a1 * f; r2 = v2 - a2 * f;
    }
    long base = (long)NTOK * 128 + (long)tok * 192 + o * 3;
    out[base] = r0; out[base + 1] = r1; out[base + 2] = r2;
  }
}

// ---------------------------------------------------------------------------
extern "C" void kernel_launch(void* const* d_in, const int* in_sizes, int n_in,
                              void* d_out, int out_size, void* d_ws, size_t ws_size,
                              hipStream_t stream) {
  (void)in_sizes; (void)n_in; (void)out_size; (void)ws_size;
  const float* x_sca    = (const float*)d_in[0];
  const float* x_vec    = (const float*)d_in[1];
  const float* edge_sca = (const float*)d_in[2];
  const float* edge_vec = (const float*)d_in[3];
  const int*   mask     = (const int*)d_in[4];
  const float* qkv_W1 = (const float*)d_in[5];
  const float* qkv_W2 = (const float*)d_in[6];
  const float* qkv_Wg = (const float*)d_in[7];
  const float* qkv_bg = (const float*)d_in[8];
  const float* qkv_Ws = (const float*)d_in[9];
  const float* qkv_bs = (const float*)d_in[10];
  const float* ek_W1  = (const float*)d_in[11];
  const float* ek_W2  = (const float*)d_in[12];
  const float* ek_Wg  = (const float*)d_in[13];
  const float* ek_bg  = (const float*)d_in[14];
  const float* ek_Ws  = (const float*)d_in[15];
  const float* ek_bs  = (const float*)d_in[16];
  const float* out_W1 = (const float*)d_in[17];
  const float* out_W2 = (const float*)d_in[18];
  const float* out_Wg = (const float*)d_in[19];
  const float* out_bg = (const float*)d_in[20];
  const float* out_Ws = (const float*)d_in[21];
  const float* out_bs = (const float*)d_in[22];
  const float* out_act_W = (const float*)d_in[23];

  float* ws = (float*)d_ws;
  float* sca_qkv = ws;                                   // 768*384
  float* vec_qkv = sca_qkv + (long)NTOK * 384;           // 768*576
  float* att_s   = vec_qkv + (long)NTOK * 576;           // 2*384*384*8
  float* att_v   = att_s + (long)BN * NN * NN * 8;
  float* so_mid  = att_v + (long)BN * NN * NN * 8;       // 768*128
  float* vo_mid  = so_mid + (long)NTOK * 128;            // 768*192

  node_qkv_kernel<<<NTOK, 128, 0, stream>>>(x_sca, x_vec, qkv_W1, qkv_W2, qkv_Wg,
                                            qkv_bg, qkv_Ws, qkv_bs, sca_qkv, vec_qkv);
  edge_att_kernel<<<BN * NN * NTILE, 32, 0, stream>>>(edge_sca, edge_vec, sca_qkv, vec_qkv,
                                            ek_W1, ek_W2, ek_Wg, ek_bg, ek_Ws, ek_bs,
                                            att_s, att_v);
  softmax_kernel<<<NTOK, 128, 0, stream>>>(att_s, att_v, mask);
  apply_att_kernel<<<NTOK, 128, 0, stream>>>(att_s, att_v, sca_qkv, vec_qkv, so_mid, vo_mid);
  out_kernel<<<NTOK, 128, 0, stream>>>(so_mid, vo_mid, out_W1, out_W2, out_Wg,
                                       out_bg, out_Ws, out_bs, out_act_W, (float*)d_out);
}